// SolveLaplace_19258633355683
// MI455X (gfx1250) — compile-verified
//
#include <hip/hip_runtime.h>

// ---------------------------------------------------------------------------
// CG solve of (Laplacian + ones) x = b on a 96x96 lattice, 16 RHS.
// Rank-one ones-matrix removed analytically (mean deflation); CG runs in the
// mean-zero subspace. Per-RHS dot products computed as 16x16 Gram matrices
// via V_WMMA_F32_16X16X4_F32 (B=16 == WMMA N dimension).
// All scalar (alpha/beta/rho) reductions are fused into the field kernels as
// redundant per-WG fixed-order reductions -> 2 launches per CG iteration.
// ---------------------------------------------------------------------------

#define Hh 96
#define Ww 96
#define Nn (Hh * Ww)          // 9216 nodes
#define Bb 16                 // batches / RHS
#define NG 144                // field workgroups; 64 nodes (4 tiles) per WG
#define TILES 4
#define ITERS 256             // fixed CG iteration count

typedef __attribute__((ext_vector_type(2))) float v2f;
typedef __attribute__((ext_vector_type(8))) float v8f;

// workspace layout (float offsets)
#define NB         (Nn * Bb)
#define OFF_C      0
#define OFF_DIAG   (OFF_C + Nn)
#define OFF_R      (OFF_DIAG + Nn)
#define OFF_P0     (OFF_R + NB)
#define OFF_P1     (OFF_P0 + NB)
#define OFF_AP     (OFF_P1 + NB)
#define OFF_X      (OFF_AP + NB)
#define OFF_GR0    (OFF_X + NB)            // r^T r Gram partials, even iters
#define OFF_GR1    (OFF_GR0 + NG * 256)    // r^T r Gram partials, odd iters
#define OFF_GPAP   (OFF_GR1 + NG * 256)    // p^T Ap Gram partials
#define OFF_COLPIN (OFF_GPAP + NG * 256)   // input column-sum partials
#define OFF_COLPX  (OFF_COLPIN + NG * 16)  // x column-sum partials

// ---- wave-level 16x16 Gram accumulation over a 16-node LDS tile ------------
// ldsA/ldsB: [16 nodes][16 batches]. acc += A_chunk^T * B_chunk over four
// K=4 slabs using V_WMMA_F32_16X16X4_F32.
// f32 16x4 A layout: lanes 0-15 hold K=0,1 in v.x,v.y; lanes 16-31 K=2,3.
__device__ inline v8f wave_gram(const float* ldsA, const float* ldsB, v8f acc) {
  int lane = threadIdx.x & 31;
  int m = lane & 15;
  int half = lane >> 4;
#pragma unroll
  for (int kc = 0; kc < 4; ++kc) {
    int k0 = kc * 4 + half * 2;
    v2f a, b;
    a.x = ldsA[(k0 + 0) * 16 + m];
    a.y = ldsA[(k0 + 1) * 16 + m];
    b.x = ldsB[(k0 + 0) * 16 + m];
    b.y = ldsB[(k0 + 1) * 16 + m];
    acc = __builtin_amdgcn_wmma_f32_16x16x4_f32(false, a, false, b,
                                                (short)0, acc, false, false);
  }
  return acc;
}

// store 16x16 accumulator per C/D layout: lane l, vgpr v -> (M=v+8*half, N=l&15)
__device__ inline void store_gram(float* gbuf, v8f acc) {
  int t = threadIdx.x;
  if (t < 32) {
    int m = t & 15, half = t >> 4;
    float* gp = gbuf + blockIdx.x * 256;
#pragma unroll
    for (int v = 0; v < 8; ++v) gp[(v + half * 8) * 16 + m] = acc[v];
  }
}

// fixed-order reduction of Gram diagonals over all NG partials -> out16[b]
__device__ inline void reduce_diag(const float* gbuf, float* red, float* out16) {
  int t = threadIdx.x, b = t & 15, ch = t >> 4;
  float s = 0.f;
  for (int w = ch; w < NG; w += 16) s += gbuf[w * 256 + b * 17];  // b*16+b
  red[t] = s;
  __syncthreads();
  if (t < 16) {
    float tot = 0.f;
    for (int k = 0; k < 16; ++k) tot += red[k * 16 + t];
    out16[t] = tot;
  }
  __syncthreads();
}

// fixed-order reduction of column-sum partials (NG x 16) -> out16[b]
__device__ inline void reduce_colp(const float* colp, float* red, float* out16) {
  int t = threadIdx.x, b = t & 15, ch = t >> 4;
  float s = 0.f;
  for (int w = ch; w < NG; w += 16) s += colp[w * 16 + b];
  red[t] = s;
  __syncthreads();
  if (t < 16) {
    float tot = 0.f;
    for (int k = 0; k < 16; ++k) tot += red[k * 16 + t];
    out16[t] = tot;
  }
  __syncthreads();
}

// ---- setup -----------------------------------------------------------------

// conductances c = exp(log_r) + input column-sum partials (input layout [B][N])
__global__ __launch_bounds__(256) void k_pre(const float* __restrict__ in,
                                             const float* __restrict__ lr,
                                             float* __restrict__ ws) {
  __shared__ float sm[256];
  int t = threadIdx.x, b = t & 15, nl = t >> 4;
  int base = blockIdx.x * (16 * TILES);
  if (t < 16 * TILES) {
    int n = base + t;
    ws[OFF_C + n] = expf(lr[n]);
  }
  float s = 0.f;
#pragma unroll
  for (int g = 0; g < TILES; ++g) s += in[b * Nn + base + g * 16 + nl];
  sm[t] = s;
  __syncthreads();
  if (t < 16) {
    float tot = 0.f;
    for (int k = 0; k < 16; ++k) tot += sm[k * 16 + t];
    ws[OFF_COLPIN + blockIdx.x * 16 + t] = tot;
  }
}

// per-WG means, Laplacian diagonal, r = in - colmean, x=0, p0=0, Gram(r,r)->GR0
__global__ __launch_bounds__(256) void k_init(const float* __restrict__ in,
                                              float* __restrict__ ws) {
  __shared__ float red[256];
  __shared__ float ms[16];
  __shared__ float ldsr[256];
  const float* c = ws + OFF_C;
  int t = threadIdx.x, b = t & 15, nl = t >> 4;
  int base = blockIdx.x * (16 * TILES);

  reduce_colp(ws + OFF_COLPIN, red, ms);   // column sums
  if (t < 16) ms[t] = ms[t] / (float)Nn;   // -> column means
  __syncthreads();

  if (t < 16 * TILES) {  // diagonal of Z for this WG's nodes
    int n = base + t;
    int row = n / Ww, col = n - row * Ww;
    float cn = c[n], d = 0.f;
    if (col > 0)      d += cn + c[n - 1];
    if (col < Ww - 1) d += cn + c[n + 1];
    if (row > 0)      d += cn + c[n - Ww];
    if (row < Hh - 1) d += cn + c[n + Ww];
    ws[OFF_DIAG + n] = d;
  }

  v8f acc = {};
  for (int g = 0; g < TILES; ++g) {
    int n = base + g * 16 + nl;
    float rv = in[b * Nn + n] - ms[b];
    ws[OFF_R + n * 16 + b] = rv;
    ws[OFF_P0 + n * 16 + b] = 0.0f;
    ws[OFF_X + n * 16 + b] = 0.0f;
    ldsr[t] = rv;
    __syncthreads();
    if (t < 32) acc = wave_gram(ldsr, ldsr, acc);
    __syncthreads();
  }
  store_gram(ws + OFF_GR0, acc);
}

// ---- CG iteration (2 kernels) ----------------------------------------------

// beta from Gram partials; p_new = r + beta*p_old; Ap = Z p_new; Gram(p,Ap)
__global__ __launch_bounds__(256) void k_stepA(float* __restrict__ ws, int it) {
  __shared__ float red[256];
  __shared__ float rc[16], rp[16];
  __shared__ float lds_pn[256], lds_ap[256];
  int par = it & 1;
  const float* gramCur = ws + (par ? OFF_GR1 : OFF_GR0);
  const float* gramPrev = ws + (par ? OFF_GR0 : OFF_GR1);
  const float* c = ws + OFF_C;
  const float* dg = ws + OFF_DIAG;
  const float* r = ws + OFF_R;
  const float* pOld = ws + (par ? OFF_P1 : OFF_P0);
  float* pNew = ws + (par ? OFF_P0 : OFF_P1);
  float* ap = ws + OFF_AP;

  reduce_diag(gramCur, red, rc);                 // rho_current
  if (it > 0) reduce_diag(gramPrev, red, rp);    // rho_previous (uniform branch)

  int t = threadIdx.x, b = t & 15, nl = t >> 4;
  float beta = 0.f;
  if (it > 0) {
    float d = rp[b];
    beta = (d != 0.f) ? rc[b] / d : 0.f;
  }
  int base = blockIdx.x * (16 * TILES);
  v8f acc = {};
  for (int g = 0; g < TILES; ++g) {
    int n = base + g * 16 + nl;
    int row = n / Ww, col = n - row * Ww;
    float cn = c[n];
    float pn = r[n * 16 + b] + beta * pOld[n * 16 + b];
    float s = dg[n] * pn;
    if (col > 0)      { int j = n - 1;  s -= (cn + c[j]) * (r[j * 16 + b] + beta * pOld[j * 16 + b]); }
    if (col < Ww - 1) { int j = n + 1;  s -= (cn + c[j]) * (r[j * 16 + b] + beta * pOld[j * 16 + b]); }
    if (row > 0)      { int j = n - Ww; s -= (cn + c[j]) * (r[j * 16 + b] + beta * pOld[j * 16 + b]); }
    if (row < Hh - 1) { int j = n + Ww; s -= (cn + c[j]) * (r[j * 16 + b] + beta * pOld[j * 16 + b]); }
    pNew[n * 16 + b] = pn;
    ap[n * 16 + b] = s;
    lds_pn[t] = pn;
    lds_ap[t] = s;
    __syncthreads();
    if (t < 32) acc = wave_gram(lds_pn, lds_ap, acc);
    __syncthreads();
  }
  store_gram(ws + OFF_GPAP, acc);
}

// alpha from Gram partials; x += alpha*p; r -= alpha*Ap; Gram(r,r) -> next slot
__global__ __launch_bounds__(256) void k_stepB(float* __restrict__ ws, int it) {
  __shared__ float red[256];
  __shared__ float rc[16], pp[16];
  __shared__ float ldsr[256];
  int par = it & 1;
  const float* gramCur = ws + (par ? OFF_GR1 : OFF_GR0);
  float* gramNext = ws + (par ? OFF_GR0 : OFF_GR1);
  const float* p = ws + (par ? OFF_P0 : OFF_P1);  // the p written by k_stepA
  const float* ap = ws + OFF_AP;
  float* x = ws + OFF_X;
  float* r = ws + OFF_R;

  reduce_diag(ws + OFF_GPAP, red, pp);  // p^T Ap
  reduce_diag(gramCur, red, rc);        // rho_current

  int t = threadIdx.x, b = t & 15, nl = t >> 4;
  float d = pp[b];
  float al = (d != 0.f) ? rc[b] / d : 0.f;
  int base = blockIdx.x * (16 * TILES);
  v8f acc = {};
  for (int g = 0; g < TILES; ++g) {
    int i = (base + g * 16 + nl) * 16 + b;
    x[i] += al * p[i];
    float rv = r[i] - al * ap[i];
    r[i] = rv;
    ldsr[t] = rv;
    __syncthreads();
    if (t < 32) acc = wave_gram(ldsr, ldsr, acc);
    __syncthreads();
  }
  store_gram(gramNext, acc);
}

// ---- finalization ----------------------------------------------------------

// partial column sums of x (layout [N][16])
__global__ __launch_bounds__(256) void k_post(float* __restrict__ ws) {
  __shared__ float sm[256];
  int t = threadIdx.x, b = t & 15, nl = t >> 4;
  int base = blockIdx.x * (16 * TILES);
  float s = 0.f;
#pragma unroll
  for (int g = 0; g < TILES; ++g) s += ws[OFF_X + (base + g * 16 + nl) * 16 + b];
  sm[t] = s;
  __syncthreads();
  if (t < 16) {
    float tot = 0.f;
    for (int k = 0; k < 16; ++k) tot += sm[k * 16 + t];
    ws[OFF_COLPX + blockIdx.x * 16 + t] = tot;
  }
}

// shift_b = (m_b - g)/N - mean(x_b); out = x + shift + offset ([N][16]->[B][N])
__global__ __launch_bounds__(256) void k_final(const float* __restrict__ ws,
                                               const float* __restrict__ off,
                                               float* __restrict__ out) {
  __shared__ float red[256];
  __shared__ float xs[16], ms[16], shift[16];
  int t = threadIdx.x;
  reduce_colp(ws + OFF_COLPX, red, xs);   // x column sums
  reduce_colp(ws + OFF_COLPIN, red, ms);  // input column sums
  if (t < 16) {
    float m = ms[t] / (float)Nn;
    ms[t] = m;
  }
  __syncthreads();
  if (t == 0) {
    float g = 0.f;
    for (int k = 0; k < 16; ++k) g += ms[k];
    red[0] = g / 16.0f;  // reuse red[0] as global mean
  }
  __syncthreads();
  if (t < 16) shift[t] = (ms[t] - red[0]) / (float)Nn - xs[t] / (float)Nn;
  __syncthreads();

  int idx = blockIdx.x * 256 + t;
  if (idx < Nn * Bb) {
    int b = idx / Nn;
    int n = idx - b * Nn;
    out[idx] = ws[OFF_X + n * 16 + b] + shift[b] + off[0];
  }
}

// ---------------------------------------------------------------------------

extern "C" void kernel_launch(void* const* d_in, const int* in_sizes, int n_in,
                              void* d_out, int out_size, void* d_ws, size_t ws_size,
                              hipStream_t stream) {
  const float* in = (const float*)d_in[0];   // [16, 96, 96]
  const float* lr = (const float*)d_in[1];   // [96, 96]
  const float* off = (const float*)d_in[2];  // scalar
  float* out = (float*)d_out;
  float* ws = (float*)d_ws;

  k_pre<<<NG, 256, 0, stream>>>(in, lr, ws);
  k_init<<<NG, 256, 0, stream>>>(in, ws);

  for (int it = 0; it < ITERS; ++it) {
    k_stepA<<<NG, 256, 0, stream>>>(ws, it);
    k_stepB<<<NG, 256, 0, stream>>>(ws, it);
  }

  k_post<<<NG, 256, 0, stream>>>(ws);
  k_final<<<(Nn * Bb + 255) / 256, 256, 0, stream>>>(ws, off, out);
}